// Voxels_22402549416458
// MI455X (gfx1250) — compile-verified
//
#include <hip/hip_runtime.h>
#include <math.h>

// Voxel lookup: pure bandwidth-bound gather. No matmul structure -> no WMMA.
// Strategy: b128-vectorized IO, EXEC-masked gather skip for outside points,
// non-temporal b128 stores (preserve L2 for the 268MB voxel grid),
// gfx1250 global_prefetch on the streaming xyz input.

typedef __attribute__((ext_vector_type(4))) float v4f;

#define NBVOX     256
#define HALFSZ    1.0f      // SCALE / 2
#define IDX_SCALE 128.0f    // NBVOX / SCALE

__device__ __forceinline__ float fast_sigmoid(float v) {
    // 1 / (1 + e^-v): v_exp_f32 + v_rcp_f32 (memory-bound kernel; 1-ulp rcp is fine)
    return __builtin_amdgcn_rcpf(1.0f + __expf(-v));
}

__device__ __forceinline__ void eval_point(float x, float y, float z,
                                           const v4f* __restrict__ vox,
                                           float* r, float* g, float* b, float* d)
{
    const bool inside = (fabsf(x) < HALFSZ) & (fabsf(y) < HALFSZ) & (fabsf(z) < HALFSZ);
    // Outside: rgb_raw = 0 -> sigmoid = 0.5 ; dens_raw = 0 -> relu = 0.
    float rr = 0.5f, gg = 0.5f, bb = 0.5f, dd = 0.0f;
    if (inside) {
        // (c + half) * nb / SCALE, truncated toward zero (matches astype(int32)),
        // clipped to [0, 255].
        int ix = (int)((x + HALFSZ) * IDX_SCALE);
        int iy = (int)((y + HALFSZ) * IDX_SCALE);
        int iz = (int)((z + HALFSZ) * IDX_SCALE);
        ix = min(max(ix, 0), NBVOX - 1);
        iy = min(max(iy, 0), NBVOX - 1);
        iz = min(max(iz, 0), NBVOX - 1);
        const size_t lin = ((size_t)ix << 16) | ((size_t)iy << 8) | (size_t)iz;
        const v4f v = vox[lin];                  // aligned global_load_b128 gather
        rr = fast_sigmoid(v.x);
        gg = fast_sigmoid(v.y);
        bb = fast_sigmoid(v.z);
        dd = fmaxf(v.w * 10.0f, 0.0f);
    }
    *r = rr; *g = gg; *b = bb; *d = dd;
}

__global__ __launch_bounds__(256)
void voxel_gather_kernel(const float* __restrict__ xyz,
                         const v4f*  __restrict__ vox,
                         float* __restrict__ rgb,
                         float* __restrict__ dens,
                         int n)
{
    const int stride = gridDim.x * blockDim.x * 4;   // points per grid sweep
    for (int base = (blockIdx.x * blockDim.x + threadIdx.x) * 4; base < n; base += stride) {
        // gfx1250 prefetch of next sweep's xyz chunk (streaming input).
        if (base + stride < n)
            __builtin_prefetch(xyz + (size_t)(base + stride) * 3, 0, 0);

        if (base + 4 <= n) {
            // 4 points = 12 floats of xyz = 3 aligned b128 loads (base % 4 == 0).
            const v4f* __restrict__ xin = (const v4f*)(xyz + (size_t)base * 3);
            const v4f a = xin[0], b4 = xin[1], c4 = xin[2];
            const float px[4] = {a.x, a.w, b4.z, c4.y};
            const float py[4] = {a.y, b4.x, b4.w, c4.z};
            const float pz[4] = {a.z, b4.y, c4.x, c4.w};

            float ro[12];
            float dv[4];
#pragma unroll
            for (int k = 0; k < 4; ++k)
                eval_point(px[k], py[k], pz[k], vox,
                           &ro[3 * k], &ro[3 * k + 1], &ro[3 * k + 2], &dv[k]);

            const v4f s0 = {ro[0], ro[1], ro[2],  ro[3]};
            const v4f s1 = {ro[4], ro[5], ro[6],  ro[7]};
            const v4f s2 = {ro[8], ro[9], ro[10], ro[11]};
            const v4f sd = {dv[0], dv[1], dv[2],  dv[3]};

            // Streaming output: non-temporal b128 stores, keep L2 for voxel gathers.
            v4f* __restrict__ rout = (v4f*)(rgb + (size_t)base * 3);
            __builtin_nontemporal_store(s0, rout + 0);
            __builtin_nontemporal_store(s1, rout + 1);
            __builtin_nontemporal_store(s2, rout + 2);
            __builtin_nontemporal_store(sd, (v4f*)(dens + base));
        } else {
            // Scalar tail (N = 4M is divisible by 4, but stay general).
            for (int p = base; p < n; ++p) {
                const float x = xyz[(size_t)p * 3 + 0];
                const float y = xyz[(size_t)p * 3 + 1];
                const float z = xyz[(size_t)p * 3 + 2];
                float r, g, b, d;
                eval_point(x, y, z, vox, &r, &g, &b, &d);
                rgb[(size_t)p * 3 + 0] = r;
                rgb[(size_t)p * 3 + 1] = g;
                rgb[(size_t)p * 3 + 2] = b;
                dens[p] = d;
            }
        }
    }
}

extern "C" void kernel_launch(void* const* d_in, const int* in_sizes, int n_in,
                              void* d_out, int out_size, void* d_ws, size_t ws_size,
                              hipStream_t stream) {
    (void)n_in; (void)out_size; (void)d_ws; (void)ws_size;

    const float* xyz = (const float*)d_in[0];          // [N, 3] f32
    const v4f*   vox = (const v4f*)d_in[1];            // [256,256,256,4] f32
    const int    n   = in_sizes[0] / 3;                // 4,000,000

    float* out  = (float*)d_out;
    float* rgb  = out;                                 // [N, 3] flat
    float* dens = out + (size_t)3 * n;                 // [N] flat (16B-aligned offset)

    // Grid-stride: ~2 sweeps/thread so the xyz prefetch has a "next" chunk.
    const int threads = 256;                           // 8 wave32 per block
    const int blocks  = 2048;
    voxel_gather_kernel<<<blocks, threads, 0, stream>>>(xyz, vox, rgb, dens, n);
}